// SeparationPrimalDualIPM_66400194396376
// MI455X (gfx1250) — compile-verified
//
#include <hip/hip_runtime.h>

// ---------------------------------------------------------------------------
// Batched primal-dual IPM for MI455X (gfx1250, wave32).
//   - 16 problems per wave; per-problem state mirrored in lanes p and p+16.
//   - Batched mat-vecs via V_WMMA_F32_16X16X4_F32 (full fp32 matrix pipe).
//   - KKT solve via diagonal-Q Schur complement + tridiagonal Thomas solve.
// ---------------------------------------------------------------------------

typedef __attribute__((ext_vector_type(2))) float v2f;
typedef __attribute__((ext_vector_type(8))) float v8f;

#define NDIM   12
#define MCON   11
#define KOUTER 15
#define KFEAS  10
#define KBACK  10
#define BETA1  0.55f
#define BETA2  0.055f

template <int LEN>
__device__ __forceinline__ float padget(const float (&x)[LEN], int k) {
    float v = x[(k < LEN) ? k : 0];
    return (k < LEN) ? v : 0.0f;
}

// D(16x16) = Amat(16xK) * X(Kx16), K = LEN (<=12), done as 3 accumulating
// 16x16x4 f32 WMMAs.  X column p = problem p's vector, supplied mirrored in
// lanes p and p+16.  Result rows 0..OUTLEN-1 of column p are gathered into
// res[] of both lanes p and p+16 (one shfl_xor(16) per accumulator VGPR).
template <int LEN, int OUTLEN>
__device__ __forceinline__ void wmma_matvec(const v2f (&a)[3], const float (&x)[LEN],
                                            bool hi, float (&res)[OUTLEN]) {
    v8f c = {};
#pragma unroll
    for (int t = 0; t < 3; ++t) {
        v2f b;
        b.x = hi ? padget<LEN>(x, 4 * t + 2) : padget<LEN>(x, 4 * t + 0);
        b.y = hi ? padget<LEN>(x, 4 * t + 3) : padget<LEN>(x, 4 * t + 1);
        c = __builtin_amdgcn_wmma_f32_16x16x4_f32(
                /*neg_a=*/false, a[t], /*neg_b=*/false, b,
                /*c_mod=*/(short)0, c, /*reuse_a=*/false, /*reuse_b=*/false);
    }
    float part[8];
#pragma unroll
    for (int j = 0; j < 8; ++j) part[j] = __shfl_xor(c[j], 16, 32);
#pragma unroll
    for (int j = 0; j < 8; ++j)
        if (j < OUTLEN) res[j] = hi ? part[j] : c[j];
#pragma unroll
    for (int j = 8; j < 16; ++j)
        if (j < OUTLEN) res[j] = hi ? c[j - 8] : part[j - 8];
}

__global__ __launch_bounds__(256) void ipm_kernel(
    const float* __restrict__ Mu, const float* __restrict__ Q,
    const float* __restrict__ Aptr, const float* __restrict__ S0,
    const float* __restrict__ Lambda, const float* __restrict__ alpha,
    float* __restrict__ out, int nprob)
{
    const int lane = threadIdx.x & 31;
    const int wave = threadIdx.x >> 5;
    const bool hi  = lane >= 16;
    const int p    = lane & 15;
    int prob = blockIdx.x * 128 + wave * 16 + p;
    if (prob >= nprob) prob = nprob - 1;            // keep EXEC full for WMMA

    __builtin_prefetch(&Mu[(size_t)prob * NDIM], 0, 3);
    __builtin_prefetch(&S0[(size_t)prob * NDIM], 0, 3);

    // ---- per-problem loads (mirrored in both lane halves) ----
    float mu[NDIM], qd[NDIM], qinv[NDIM], s[NDIM];
#pragma unroll
    for (int n = 0; n < NDIM; ++n) {
        mu[n]   = Mu[(size_t)prob * NDIM + n];
        s[n]    = S0[(size_t)prob * NDIM + n];
        qd[n]   = Q[(size_t)prob * NDIM * NDIM + n * (NDIM + 1)];
        qinv[n] = 1.0f / qd[n];
    }
    float lam[MCON];
#pragma unroll
    for (int m = 0; m < MCON; ++m) lam[m] = Lambda[(size_t)prob * MCON + m];
    const float alp = alpha[prob];

    // ---- WMMA operands: A (11x12) and A^T (12x11), zero-padded to 16x12 ----
    v2f aA[3], aT[3];
    {
        const int r = lane & 15;
        float arow[NDIM];
#pragma unroll
        for (int n = 0; n < NDIM; ++n)
            arow[n] = (r < MCON) ? Aptr[r * NDIM + n] : 0.0f;
        float acol[NDIM];                 // A^T row r = column r of A; pad K=11
#pragma unroll
        for (int m = 0; m < MCON; ++m)
            acol[m] = (r < NDIM) ? Aptr[m * NDIM + r] : 0.0f;
        acol[MCON] = 0.0f;
#pragma unroll
        for (int t = 0; t < 3; ++t) {
            aA[t].x = hi ? arow[4 * t + 2] : arow[4 * t + 0];
            aA[t].y = hi ? arow[4 * t + 3] : arow[4 * t + 1];
            aT[t].x = hi ? acol[4 * t + 2] : acol[4 * t + 0];
            aT[t].y = hi ? acol[4 * t + 3] : acol[4 * t + 1];
        }
    }

    // ---- iteration-invariant Schur tridiagonal G = A Qinv A^T ----
    float gd[MCON], ge[MCON - 1];
#pragma unroll
    for (int i = 0; i < MCON; ++i) gd[i] = qinv[i] + qinv[i + 1];
#pragma unroll
    for (int i = 0; i < MCON - 1; ++i) ge[i] = -qinv[i + 1];

    float as_[MCON];
    wmma_matvec<NDIM, MCON>(aA, s, hi, as_);        // AS = A @ S0

#pragma unroll 1
    for (int it = 0; it < KOUTER; ++it) {
        // t = -alpha*M / (AS^T Lam)
        float dot = 0.0f;
#pragma unroll
        for (int m = 0; m < MCON; ++m) dot += as_[m] * lam[m];
        const float invt = dot / (-alp * (float)MCON);

        // residual R = [R1; R2]
        float atl[NDIM];
        wmma_matvec<MCON, NDIM>(aT, lam, hi, atl);  // A^T @ Lam
        float R1[NDIM], R2[MCON];
#pragma unroll
        for (int n = 0; n < NDIM; ++n) R1[n] = qd[n] * (s[n] - mu[n]) + atl[n];
#pragma unroll
        for (int m = 0; m < MCON; ++m) R2[m] = -lam[m] * as_[m] - invt;

        // Schur RHS: rhs = R2 + Lam .* (A Qinv R1)   (A bidiagonal here)
        float v[NDIM], rhs[MCON];
#pragma unroll
        for (int n = 0; n < NDIM; ++n) v[n] = qinv[n] * R1[n];
#pragma unroll
        for (int m = 0; m < MCON; ++m) rhs[m] = R2[m] + lam[m] * (v[m] - v[m + 1]);

        // Thomas solve:  [diag(Lam)G - diag(AS)] xL = rhs
        float cc[MCON], yy[MCON];
        {
            float d0 = lam[0] * gd[0] - as_[0];
            float i0 = 1.0f / d0;
            cc[0] = (lam[0] * ge[0]) * i0;
            yy[0] = rhs[0] * i0;
#pragma unroll
            for (int i = 1; i < MCON; ++i) {
                float li  = lam[i] * ge[i - 1];
                float di  = lam[i] * gd[i] - as_[i] - li * cc[i - 1];
                float inv = 1.0f / di;
                cc[i] = ((i < MCON - 1) ? lam[i] * ge[i] : 0.0f) * inv;
                yy[i] = (rhs[i] - li * yy[i - 1]) * inv;
            }
        }
        float xL[MCON];
        xL[MCON - 1] = yy[MCON - 1];
#pragma unroll
        for (int i = MCON - 2; i >= 0; --i) xL[i] = yy[i] - cc[i] * xL[i + 1];

        // back-substitute primal direction
        float pds[NDIM], pdl[MCON];
#pragma unroll
        for (int n = 0; n < NDIM; ++n) {
            float atx = ((n < MCON) ? xL[n] : 0.0f) - ((n >= 1) ? xL[n - 1] : 0.0f);
            pds[n] = -(qinv[n] * (R1[n] - atx));
        }
#pragma unroll
        for (int m = 0; m < MCON; ++m) pdl[m] = -xL[m];

        // initial step = 0.99 * min(-Lam / negPDL)
        float mn = 3.4e38f;
#pragma unroll
        for (int m = 0; m < MCON; ++m) {
            float np = (pdl[m] < 0.0f) ? pdl[m] : -lam[m];
            mn = fminf(mn, -lam[m] / np);
        }
        float step = 0.99f * mn;

        float apds[MCON];
        wmma_matvec<NDIM, MCON>(aA, pds, hi, apds); // A @ PD_S

        // feasibility backtracking (branchless)
#pragma unroll 1
        for (int f = 0; f < KFEAS; ++f) {
            bool viol = false;
#pragma unroll
            for (int m = 0; m < MCON; ++m) viol = viol || (as_[m] + step * apds[m] > 0.0f);
            step = viol ? step * BETA1 : step;
        }

        float rn2 = 0.0f;
#pragma unroll
        for (int n = 0; n < NDIM; ++n) rn2 += R1[n] * R1[n];
#pragma unroll
        for (int m = 0; m < MCON; ++m) rn2 += R2[m] * R2[m];
        const float RNorm = sqrtf(rn2);

        float atpdl[NDIM];
        wmma_matvec<MCON, NDIM>(aT, pdl, hi, atpdl); // A^T @ PD_L

        // residual backtracking (branchless)
#pragma unroll 1
        for (int bk = 0; bk < KBACK; ++bk) {
            float r2s = 0.0f;
#pragma unroll
            for (int n = 0; n < NDIM; ++n) {
                float r1 = qd[n] * (s[n] + step * pds[n] - mu[n]) + atl[n] + step * atpdl[n];
                r2s += r1 * r1;
            }
#pragma unroll
            for (int m = 0; m < MCON; ++m) {
                float r2 = -(lam[m] + step * pdl[m]) * (as_[m] + step * apds[m]) - invt;
                r2s += r2 * r2;
            }
            step = (sqrtf(r2s) > (1.0f - BETA2 * step) * RNorm) ? step * BETA1 : step;
        }

        // update
#pragma unroll
        for (int n = 0; n < NDIM; ++n) s[n] += step * pds[n];
#pragma unroll
        for (int m = 0; m < MCON; ++m) lam[m] += step * pdl[m];
        wmma_matvec<NDIM, MCON>(aA, s, hi, as_);    // AS = A @ S_new
    }

    if (!hi) {
#pragma unroll
        for (int n = 0; n < NDIM; ++n) out[(size_t)prob * NDIM + n] = s[n];
    }
}

extern "C" void kernel_launch(void* const* d_in, const int* in_sizes, int n_in,
                              void* d_out, int out_size, void* d_ws, size_t ws_size,
                              hipStream_t stream) {
    const float* Mu  = (const float*)d_in[0];
    const float* Q   = (const float*)d_in[1];
    const float* A   = (const float*)d_in[2];
    const float* S0  = (const float*)d_in[3];
    const float* Lam = (const float*)d_in[4];
    const float* al  = (const float*)d_in[5];
    const int nprob  = in_sizes[5];                 // B*W (alpha count)
    const int blocks = (nprob + 127) / 128;         // 8 waves/block * 16 problems/wave
    ipm_kernel<<<blocks, 256, 0, stream>>>(Mu, Q, A, S0, Lam, al, (float*)d_out, nprob);
}